// Model_70471823392995
// MI455X (gfx1250) — compile-verified
//
#include <hip/hip_runtime.h>
#include <stdint.h>

// Pack per-row top-k expert ids (8 ids in [0,128) per row) into a 128-bit
// bitmask stored as 4 x uint32 per row. Pure bandwidth-bound bit-packing:
// 32 B in + 16 B out per row. One thread per row, b128 loads/stores.

typedef uint32_t v4u __attribute__((ext_vector_type(4)));

__global__ __launch_bounds__(256) void topk_bitpack_kernel(
    const v4u* __restrict__ in,   // topk_ids viewed as 2 x v4u per row
    v4u* __restrict__ out,        // bitmatrix: 1 v4u per row
    int n_rows) {
  int r = blockIdx.x * 256 + threadIdx.x;
  if (r >= n_rows) return;

  // gfx1250 prefetch path (global_prefetch_b8): no counters, no VGPR return.
  __builtin_prefetch(&in[2 * r], 0, 0);

  // Read-once input: non-temporal hint (TH_LOAD_NT) keeps it from churning L2.
  v4u a = __builtin_nontemporal_load(&in[2 * r + 0]);
  v4u b = __builtin_nontemporal_load(&in[2 * r + 1]);

  uint64_t lo = 0ull;  // bits 0..63   (experts 0..63)
  uint64_t hi = 0ull;  // bits 64..127 (experts 64..127)

  uint32_t id;
  // Fully unrolled; 128-bit mask built as two u64 halves.
  // m = (id >= 64) ? bit : 0 ; hi |= m ; lo |= bit ^ m  (XOR avoids 2nd select)
#define ACC(IDV)                                   \
  do {                                             \
    id = (uint32_t)(IDV) & 127u;                   \
    uint64_t bit = 1ull << (id & 63u);             \
    uint64_t m = (id & 64u) ? bit : 0ull;          \
    hi |= m;                                       \
    lo |= bit ^ m;                                 \
  } while (0)

  ACC(a.x); ACC(a.y); ACC(a.z); ACC(a.w);
  ACC(b.x); ACC(b.y); ACC(b.z); ACC(b.w);
#undef ACC

  v4u o;
  o.x = (uint32_t)(lo);
  o.y = (uint32_t)(lo >> 32);
  o.z = (uint32_t)(hi);
  o.w = (uint32_t)(hi >> 32);

  // Regular-temporal store: 16 MB result stays resident in the 192 MB L2
  // for whoever consumes the bitmatrix next.
  out[r] = o;
}

extern "C" void kernel_launch(void* const* d_in, const int* in_sizes, int n_in,
                              void* d_out, int out_size, void* d_ws, size_t ws_size,
                              hipStream_t stream) {
  (void)n_in; (void)out_size; (void)d_ws; (void)ws_size;

  // d_in[0]: topk_ids int32 [n_rows, 8]; d_in[1]: n_experts (==128, baked in).
  const v4u* in = (const v4u*)d_in[0];
  v4u* out = (v4u*)d_out;

  int n_rows = in_sizes[0] / 8;
  int blocks = (n_rows + 255) / 256;

  topk_bitpack_kernel<<<blocks, 256, 0, stream>>>(in, out, n_rows);
}